// interactionModule_50328426774889
// MI455X (gfx1250) — compile-verified
//
#include <hip/hip_runtime.h>
#include <cstdint>

// LJ edge-force scatter for MI455X (gfx1250).
//
//   m = dr * (48 r^-14 - 24 r^-8) = dr * 24 * r^-8 * (2 r^-6 - 1),  r^2 = dot(dr,dr)
//   v[dst] += m   (segment sum -> device-scope f32 atomics at L2)
//
// Edge-index stream (25.6 MB, the only real HBM traffic) is staged through LDS
// with the CDNA5 async pipeline: global_load_async_to_lds_b128 double-buffered
// one tile ahead, synchronized per-wave with s_wait_asynccnt (no barriers:
// each lane consumes exactly the 16B region it async-loaded).

__global__ void zero_out_kernel(float* __restrict__ out, int n) {
    int i = blockIdx.x * blockDim.x + threadIdx.x;
    if (i < n) out[i] = 0.0f;
}

// Fire-and-forget f32 atomic add at device scope: RMW executes at L2 (which
// holds the whole 1.2MB output), no return value -> STOREcnt only.
__device__ __forceinline__ void atomic_add_f32_dev(float* p, float v) {
    asm volatile("global_atomic_add_f32 %0, %1, off scope:SCOPE_DEV"
                 :: "v"((unsigned long long)(uintptr_t)p), "v"(v)
                 : "memory");
}

// Async copy 16B from global to LDS (GV mode, tracked by ASYNCcnt).
__device__ __forceinline__ void async_load_b128(unsigned lds_addr, const void* gptr) {
    asm volatile("global_load_async_to_lds_b128 %0, %1, off"
                 :: "v"(lds_addr), "v"((unsigned long long)(uintptr_t)gptr)
                 : "memory");
}

constexpr int BLOCK = 256;            // 8 wave32s
constexpr int EPT   = 4;              // edges per thread per tile (16B chunk)
constexpr int TILE  = BLOCK * EPT;    // 1024 edges per tile
constexpr int NT    = 4;              // tiles per block -> 4096 edges/block

__device__ __forceinline__ void do_edge(const float* __restrict__ x,
                                        float* __restrict__ out,
                                        int s, int d) {
    const float* xs = x + 3 * s;      // gathers hit the 192MB L2 (x is 1.2MB)
    const float* xd = x + 3 * d;
    const float dx = xs[0] - xd[0];
    const float dy = xs[1] - xd[1];
    const float dz = xs[2] - xd[2];

    const float r2     = fmaf(dx, dx, fmaf(dy, dy, dz * dz));
    const float inv_r2 = 1.0f / r2;                       // src != dst -> r2 > 0
    const float inv_r6 = inv_r2 * inv_r2 * inv_r2;
    const float coef   = 24.0f * inv_r6 * inv_r2 * fmaf(2.0f, inv_r6, -1.0f);

    float* o = out + 3 * d;
    atomic_add_f32_dev(o + 0, coef * dx);
    atomic_add_f32_dev(o + 1, coef * dy);
    atomic_add_f32_dev(o + 2, coef * dz);
}

__global__ void __launch_bounds__(BLOCK)
lj_edges_kernel(const float* __restrict__ x,
                const int* __restrict__ src,
                const int* __restrict__ dst,
                float* __restrict__ out, int nE) {
    __shared__ alignas(16) int sSrc[2][TILE];
    __shared__ alignas(16) int sDst[2][TILE];

    const int tid  = threadIdx.x;
    const int base = blockIdx.x * (TILE * NT);

    // Block-uniform predicate: tile fully inside the edge array -> every lane's
    // 16B async chunk is in-bounds and every wave's ASYNCcnt count is exact.
    auto tile_full = [&](int t) -> bool {
        return base + (t + 1) * TILE <= nE;
    };
    auto issue = [&](int t) {
        const int buf = t & 1;
        const int e   = base + t * TILE + tid * EPT;
        async_load_b128((unsigned)(uintptr_t)&sSrc[buf][tid * EPT], src + e);
        async_load_b128((unsigned)(uintptr_t)&sDst[buf][tid * EPT], dst + e);
    };

    if (tile_full(0)) issue(0);

    for (int t = 0; t < NT; ++t) {
        const bool curFull = tile_full(t);
        const bool nxtFull = (t + 1 < NT) && tile_full(t + 1);

        if (nxtFull) issue(t + 1);    // prefetch next tile while we compute

        if (curFull) {
            // Wait for tile t's 2 async loads; leave next tile's 2 in flight.
            if (nxtFull) asm volatile("s_wait_asynccnt 0x2" ::: "memory");
            else         asm volatile("s_wait_asynccnt 0x0" ::: "memory");

            const int buf = t & 1;
            const int4 sv = *(const int4*)&sSrc[buf][tid * EPT];  // ds_load_b128
            const int4 dv = *(const int4*)&sDst[buf][tid * EPT];
            do_edge(x, out, sv.x, dv.x);
            do_edge(x, out, sv.y, dv.y);
            do_edge(x, out, sv.z, dv.z);
            do_edge(x, out, sv.w, dv.w);
        } else {
            // Ragged tail tile: guarded non-temporal direct loads.
            const int e0 = base + t * TILE + tid * EPT;
#pragma unroll
            for (int k = 0; k < EPT; ++k) {
                const int e = e0 + k;
                if (e < nE) {
                    const int s = __builtin_nontemporal_load(&src[e]);
                    const int d = __builtin_nontemporal_load(&dst[e]);
                    do_edge(x, out, s, d);
                }
            }
        }
    }
}

extern "C" void kernel_launch(void* const* d_in, const int* in_sizes, int n_in,
                              void* d_out, int out_size, void* d_ws, size_t ws_size,
                              hipStream_t stream) {
    const float* x   = (const float*)d_in[0];   // [N, 3] positions
    const int*   src = (const int*)d_in[1];     // [E]
    const int*   dst = (const int*)d_in[2];     // [E]
    float*       out = (float*)d_out;           // [N, 3]
    const int nE = in_sizes[1];

    // Harness poisons d_out with 0xAA -> zero before accumulating.
    {
        const int threads = 256;
        const int blocks  = (out_size + threads - 1) / threads;
        zero_out_kernel<<<blocks, threads, 0, stream>>>(out, out_size);
    }

    const int edgesPerBlock = TILE * NT;
    const int grid = (nE + edgesPerBlock - 1) / edgesPerBlock;
    lj_edges_kernel<<<grid, BLOCK, 0, stream>>>(x, src, dst, out, nE);
}